// GNNModel_58961311040421
// MI455X (gfx1250) — compile-verified
//
#include <hip/hip_runtime.h>
#include <cstdint>
#include <cstddef>

#define HID 128
#define RED 8          // HID / 16-reduction = 128/16

typedef float v2f __attribute__((ext_vector_type(2)));
typedef float v8f __attribute__((ext_vector_type(8)));

// ---- helpers -------------------------------------------------------------

__device__ __forceinline__ unsigned fkey(float f) {
    // order-preserving map float -> unsigned for atomicMax
    unsigned u = __float_as_uint(f);
    return (u & 0x80000000u) ? ~u : (u | 0x80000000u);
}
__device__ __forceinline__ float fdecode(unsigned k) {
    unsigned u = (k & 0x80000000u) ? (k ^ 0x80000000u) : ~k;
    return __uint_as_float(u);
}
__device__ __forceinline__ float elu1(float v) {
    return v > 0.f ? v : (__expf(v) - 1.f);
}

// ---- generic zero --------------------------------------------------------

__global__ void zero_f32(float* __restrict__ p, size_t n) {
    size_t i = (size_t)blockIdx.x * blockDim.x + threadIdx.x;
    if (i < n) p[i] = 0.f;
}

// ---- layer-1 edge scatter (C = 3) ---------------------------------------

__global__ void scatter3(const float* __restrict__ x,
                         const int* __restrict__ row, const int* __restrict__ col,
                         const float* __restrict__ dis,
                         float* __restrict__ agg, int E) {
    int e = blockIdx.x * blockDim.x + threadIdx.x;
    if (e >= E) return;
    int r = row[e], c = col[e];
    float norm = dis[r] * dis[c];
    atomicAdd(&agg[(size_t)c * 3 + 0], x[(size_t)r * 3 + 0] * norm);
    atomicAdd(&agg[(size_t)c * 3 + 1], x[(size_t)r * 3 + 1] * norm);
    atomicAdd(&agg[(size_t)c * 3 + 2], x[(size_t)r * 3 + 2] * norm);
}

// ---- edge scatter (C = 128), gate of previous layer fused in ------------
// one wave32 per edge, 4 channels per lane (float4)

__global__ void scatter128(const float* __restrict__ h, const float* __restrict__ att,
                           const int* __restrict__ row, const int* __restrict__ col,
                           const float* __restrict__ dis,
                           float* __restrict__ agg, int E) {
    int gt = blockIdx.x * blockDim.x + threadIdx.x;
    int e = gt >> 5;
    int lane = gt & 31;
    if (e >= E) return;
    int r = row[e], c = col[e];
    float norm = dis[r] * dis[c];
    float4 v = reinterpret_cast<const float4*>(h + (size_t)r * HID)[lane];
    float4 a = reinterpret_cast<const float4*>(att)[lane];
    float* dst = agg + (size_t)c * HID + lane * 4;
    atomicAdd(dst + 0, v.x * a.x * norm);
    atomicAdd(dst + 1, v.y * a.y * norm);
    atomicAdd(dst + 2, v.z * a.z * norm);
    atomicAdd(dst + 3, v.w * a.w * norm);
}

// ---- layer-1 GEMM (K = 3) + ELU + pooled stats --------------------------
// block: 128 threads (one per channel), 64 nodes per block

__global__ void gemm_k3(const float* __restrict__ agg, const float* __restrict__ W,
                        const float* __restrict__ b, float* __restrict__ h,
                        float* __restrict__ sum, unsigned* __restrict__ maxk, int N) {
    int c = threadIdx.x;
    int n0 = blockIdx.x * 64;
    float w0 = W[c * 3 + 0], w1 = W[c * 3 + 1], w2 = W[c * 3 + 2], bb = b[c];
    float s = 0.f, mx = -3.4e38f;
    int end = (n0 + 64 < N) ? n0 + 64 : N;
    for (int n = n0; n < end; ++n) {
        float v = agg[(size_t)n * 3 + 0] * w0 + agg[(size_t)n * 3 + 1] * w1
                + agg[(size_t)n * 3 + 2] * w2 + bb;
        v = elu1(v);
        h[(size_t)n * HID + c] = v;
        s += v;
        mx = fmaxf(mx, v);
    }
    atomicAdd(&sum[c], s);
    atomicMax(&maxk[c], fkey(mx));
}

// ---- WMMA GEMM: h = elu(agg[N,128] @ W.T + b) + fused pooled stats ------
// block = 256 threads = 8 waves; each wave -> one 16x16 output tile
// A tile (16x128) staged in LDS (stride 132 avoids bank conflicts)
// W rows served from global (64KB, L2/WGP$ resident)

__global__ void __launch_bounds__(256)
gemm_wmma(const float* __restrict__ agg, const float* __restrict__ W,
          const float* __restrict__ b, float* __restrict__ h,
          float* __restrict__ sum, unsigned* __restrict__ maxk, int N) {
    __shared__ float As[16 * 132];

    int tid  = threadIdx.x;
    int wave = tid >> 5;          // 0..7 -> output-channel tile
    int lane = tid & 31;
    int nloc = lane & 15;         // column within tile / row M within tile
    int half = lane >> 4;         // 0/1
    int n0   = blockIdx.x * 16;   // node base

    // cooperative load of A tile: 16 rows x 128 k
    for (int i = tid; i < 16 * 128; i += 256) {
        int m = i >> 7;
        int k = i & 127;
        int nr = n0 + m;
        As[m * 132 + k] = (nr < N) ? agg[(size_t)nr * HID + k] : 0.f;
    }
    __syncthreads();

    int ch = wave * 16 + nloc;                 // output channel for this lane
    const float* Wrow = W + (size_t)ch * HID;  // W[ch][k]

    v8f acc = {};
#pragma unroll
    for (int kb = 0; kb < HID; kb += 4) {
        // A layout: a[j] = A[m][kb + 2*half + j]
        v2f a = *reinterpret_cast<const v2f*>(&As[nloc * 132 + kb + 2 * half]);
        // B layout: b[j] = B[kb + 2*half + j][n] = W[n][kb + 2*half + j]
        v2f bb = *reinterpret_cast<const v2f*>(&Wrow[kb + 2 * half]);
        acc = __builtin_amdgcn_wmma_f32_16x16x4_f32(
            false, a, false, bb, (short)0, acc, false, false);
    }

    // epilogue: bias + ELU + store + pooled stats
    float bias = b[ch];
    float s = 0.f, mx = -3.4e38f;
#pragma unroll
    for (int r = 0; r < 8; ++r) {
        int m = r + half * 8;     // node row within tile
        float v = acc[r] + bias;
        v = elu1(v);
        if (n0 + m < N) h[(size_t)(n0 + m) * HID + ch] = v;
        s += v;
        mx = fmaxf(mx, v);
    }
    s += __shfl_xor(s, 16, 32);
    mx = fmaxf(mx, __shfl_xor(mx, 16, 32));
    if (half == 0) {
        atomicAdd(&sum[ch], s);
        atomicMax(&maxk[ch], fkey(mx));
    }
}

// ---- channel-attention gate: att = sigmoid(mlp(avg) + mlp(max)) ---------
// single block, 128 threads

__global__ void gate_kernel(const float* __restrict__ sum, const unsigned* __restrict__ maxk,
                            const float* __restrict__ w1, const float* __restrict__ w2,
                            float* __restrict__ att, float invN) {
    __shared__ float sv[HID], sm[HID], ha[RED], hm[RED];
    int t = threadIdx.x;
    sv[t] = sum[t] * invN;
    sm[t] = fdecode(maxk[t]);
    __syncthreads();
    if (t < RED) {
        float a = 0.f, m = 0.f;
        for (int c = 0; c < HID; ++c) {
            float w = w1[t * HID + c];
            a += sv[c] * w;
            m += sm[c] * w;
        }
        ha[t] = fmaxf(a, 0.f);
        hm[t] = fmaxf(m, 0.f);
    }
    __syncthreads();
    float o = 0.f;
    for (int j = 0; j < RED; ++j) o += (ha[j] + hm[j]) * w2[t * RED + j];
    att[t] = 1.f / (1.f + __expf(-o));
}

// ---- output head: out = (h * att) @ Wout.T + bout -----------------------
// one wave per node, float4 per lane, xor-shuffle reduce

__global__ void outhead(const float* __restrict__ h, const float* __restrict__ att,
                        const float* __restrict__ Wout, const float* __restrict__ bout,
                        float* __restrict__ out, int N) {
    int gt = blockIdx.x * blockDim.x + threadIdx.x;
    int node = gt >> 5;
    int lane = gt & 31;
    if (node >= N) return;
    float4 v = reinterpret_cast<const float4*>(h + (size_t)node * HID)[lane];
    float4 a = reinterpret_cast<const float4*>(att)[lane];
    v.x *= a.x; v.y *= a.y; v.z *= a.z; v.w *= a.w;
    float acc[3];
#pragma unroll
    for (int o = 0; o < 3; ++o) {
        float4 w = reinterpret_cast<const float4*>(Wout + (size_t)o * HID)[lane];
        acc[o] = v.x * w.x + v.y * w.y + v.z * w.z + v.w * w.w;
    }
#pragma unroll
    for (int off = 16; off >= 1; off >>= 1) {
        acc[0] += __shfl_xor(acc[0], off, 32);
        acc[1] += __shfl_xor(acc[1], off, 32);
        acc[2] += __shfl_xor(acc[2], off, 32);
    }
    if (lane < 3) out[(size_t)node * 3 + lane] = acc[lane] + bout[lane];
}

// ---- host launch ---------------------------------------------------------

extern "C" void kernel_launch(void* const* d_in, const int* in_sizes, int n_in,
                              void* d_out, int out_size, void* d_ws, size_t ws_size,
                              hipStream_t stream) {
    const float* x     = (const float*)d_in[0];
    const int*   ei    = (const int*)  d_in[1];
    const float* dis   = (const float*)d_in[2];
    const float* W1    = (const float*)d_in[3];
    const float* b1    = (const float*)d_in[4];
    const float* W2    = (const float*)d_in[5];
    const float* b2    = (const float*)d_in[6];
    const float* W3    = (const float*)d_in[7];
    const float* b3    = (const float*)d_in[8];
    const float* ca1w1 = (const float*)d_in[9];
    const float* ca1w2 = (const float*)d_in[10];
    const float* ca2w1 = (const float*)d_in[11];
    const float* ca2w2 = (const float*)d_in[12];
    const float* ca3w1 = (const float*)d_in[13];
    const float* ca3w2 = (const float*)d_in[14];
    const float* Wout  = (const float*)d_in[15];
    const float* bout  = (const float*)d_in[16];

    int N = in_sizes[0] / 3;
    int E = in_sizes[1] / 2;
    const int* row = ei;
    const int* col = ei + E;

    float*    bufA = (float*)d_ws;                   // agg buffer
    float*    bufB = bufA + (size_t)N * HID;         // h buffer
    float*    sum  = bufB + (size_t)N * HID;         // [128]
    unsigned* maxk = (unsigned*)(sum + HID);         // [128]
    float*    att  = (float*)(maxk + HID);           // [128]
    float*    out  = (float*)d_out;
    float invN = 1.f / (float)N;

    unsigned zb_small = (unsigned)((256 + 255) / 256);
    unsigned zb_n3    = (unsigned)(((size_t)N * 3 + 255) / 256);
    unsigned zb_nh    = (unsigned)(((size_t)N * HID + 255) / 256);
    unsigned gb_edges = (unsigned)((E + 255) / 256);
    unsigned gb_escat = (unsigned)(((size_t)E * 32 + 255) / 256);
    unsigned gb_k3    = (unsigned)((N + 63) / 64);
    unsigned gb_wmma  = (unsigned)((N + 15) / 16);
    unsigned gb_head  = (unsigned)(((size_t)N * 32 + 255) / 256);

    // ---- layer 1 (C_in = 3) ----
    zero_f32<<<zb_n3, 256, 0, stream>>>(bufA, (size_t)N * 3);
    zero_f32<<<zb_small, 256, 0, stream>>>(sum, 256);
    scatter3<<<gb_edges, 256, 0, stream>>>(x, row, col, dis, bufA, E);
    gemm_k3<<<gb_k3, 128, 0, stream>>>(bufA, W1, b1, bufB, sum, maxk, N);
    gate_kernel<<<1, 128, 0, stream>>>(sum, maxk, ca1w1, ca1w2, att, invN);

    // ---- layer 2 ----
    zero_f32<<<zb_nh, 256, 0, stream>>>(bufA, (size_t)N * HID);
    zero_f32<<<zb_small, 256, 0, stream>>>(sum, 256);
    scatter128<<<gb_escat, 256, 0, stream>>>(bufB, att, row, col, dis, bufA, E);
    gemm_wmma<<<gb_wmma, 256, 0, stream>>>(bufA, W2, b2, bufB, sum, maxk, N);
    gate_kernel<<<1, 128, 0, stream>>>(sum, maxk, ca2w1, ca2w2, att, invN);

    // ---- layer 3 ----
    zero_f32<<<zb_nh, 256, 0, stream>>>(bufA, (size_t)N * HID);
    zero_f32<<<zb_small, 256, 0, stream>>>(sum, 256);
    scatter128<<<gb_escat, 256, 0, stream>>>(bufB, att, row, col, dis, bufA, E);
    gemm_wmma<<<gb_wmma, 256, 0, stream>>>(bufA, W3, b3, bufB, sum, maxk, N);
    gate_kernel<<<1, 128, 0, stream>>>(sum, maxk, ca3w1, ca3w2, att, invN);

    // ---- output head ----
    outhead<<<gb_head, 256, 0, stream>>>(bufB, att, Wout, bout, out, N);
}